// GNN_52527450030253
// MI455X (gfx1250) — compile-verified
//
#include <hip/hip_runtime.h>

// 2-layer GCN (PyG GCNConv semantics) for MI455X / gfx1250.
//
// Cost analysis: dense math is ~8 MFLOP (negligible); the edge passes dominate
// (~460 MB of gather + f32-atomic scatter traffic, L2-resident on a 192 MB L2).
// The x@W1 (Nx5 @ 5x16) transform uses V_WMMA_F32_16X16X4_F32 (full fp32).

typedef float v2f __attribute__((ext_vector_type(2)));
typedef float v8f __attribute__((ext_vector_type(8)));

namespace {
constexpr int kHid = 16;   // hidden width (W1 is 5x16)
constexpr int kFin = 5;    // input feature width
}

// ---------------------------------------------------------------------------
// deg[n] = 1 (self loop); then scatter-add 1 per edge onto dst; then 1/sqrt.
// ---------------------------------------------------------------------------
__global__ void k_init_deg(float* __restrict__ deg, int n) {
  int i = blockIdx.x * blockDim.x + threadIdx.x;
  if (i < n) deg[i] = 1.0f;  // self-loop contribution
}

__global__ void k_degree(const int* __restrict__ dst, float* __restrict__ deg, int e) {
  int i = blockIdx.x * blockDim.x + threadIdx.x;
  if (i < e) atomicAdd(&deg[dst[i]], 1.0f);
}

__global__ void k_rsqrt(float* __restrict__ deg, int n) {
  int i = blockIdx.x * blockDim.x + threadIdx.x;
  if (i < n) deg[i] = 1.0f / sqrtf(deg[i]);  // deg >= 1 always (self loops)
}

// ---------------------------------------------------------------------------
// h = x @ W1 via V_WMMA_F32_16X16X4_F32. One wave per 16-row tile.
// K=5 handled as two K=4 WMMA steps (K=5..7 zero padded).
// A layout (ISA 7.12.2, 32-bit A 16x4): lane&15 = M; lanes 0-15 carry K={0,1}
// in (v0,v1), lanes 16-31 carry K={2,3}. B symmetric with lane&15 = N.
// C/D: VGPR r -> M = r + 8*(lane>=16), N = lane&15.
// ---------------------------------------------------------------------------
__global__ void k_xw1_wmma(const float* __restrict__ x,
                           const float* __restrict__ W1,
                           float* __restrict__ h,
                           int n, int ntiles) {
  const int lane = threadIdx.x & 31;
  const int wave = threadIdx.x >> 5;
  const int tile = blockIdx.x * (blockDim.x >> 5) + wave;
  if (tile >= ntiles) return;  // wave-uniform: EXEC stays all-ones for WMMA

  const int half = lane >> 4;   // 0 -> K base 0 ; 1 -> K base 2
  const int idx  = lane & 15;   // M for A, N for B
  int row = tile * 16 + idx;
  if (row >= n) row = n - 1;    // clamp loads (tail tile only); stores guarded
  const float* xr = x + row * kFin;

  // K-step 0: K = 0..3
  v2f a0, b0;
  a0.x = xr[half ? 2 : 0];
  a0.y = xr[half ? 3 : 1];
  b0.x = W1[(half ? 2 : 0) * kHid + idx];
  b0.y = W1[(half ? 3 : 1) * kHid + idx];

  // K-step 1: K = 4..7 (only K=4 is real)
  v2f a1, b1;
  a1.x = half ? 0.0f : xr[4];
  a1.y = 0.0f;
  b1.x = half ? 0.0f : W1[4 * kHid + idx];
  b1.y = 0.0f;

  v8f c = {};
  c = __builtin_amdgcn_wmma_f32_16x16x4_f32(false, a0, false, b0, (short)0, c,
                                            false, false);
  c = __builtin_amdgcn_wmma_f32_16x16x4_f32(false, a1, false, b1, (short)0, c,
                                            false, false);

  // Store D. Full-tile check is wave-uniform -> no divergent exec juggling on
  // the common path (N % 16 == 0 means every tile is full).
  const int m0 = tile * 16 + half * 8;
  if (tile * 16 + 16 <= n) {
    float* ht = h + m0 * kHid + idx;
#pragma unroll
    for (int r = 0; r < 8; ++r) ht[r * kHid] = c[r];
  } else {
#pragma unroll
    for (int r = 0; r < 8; ++r) {
      int m = m0 + r;
      if (m < n) h[m * kHid + idx] = c[r];
    }
  }
}

// ---------------------------------------------------------------------------
// agg[n][f] = h[n][f] * dinv[n]^2  (self-loop message, also initializes agg)
// ---------------------------------------------------------------------------
__global__ void k_selfloop16(const float* __restrict__ h,
                             const float* __restrict__ dinv,
                             float* __restrict__ agg, int n16) {
  int i = blockIdx.x * blockDim.x + threadIdx.x;
  if (i < n16) {
    float di = dinv[i >> 4];
    agg[i] = h[i] * di * di;
  }
}

// ---------------------------------------------------------------------------
// Layer-1 edge pass: 16 lanes per edge (coalesced 64B gather + 64B atomic
// scatter). agg[d] += h[s] * dinv[s]*dinv[d]. All offsets fit in 32-bit.
// ---------------------------------------------------------------------------
__global__ void k_edge16(const int* __restrict__ src, const int* __restrict__ dst,
                         const float* __restrict__ h, const float* __restrict__ dinv,
                         float* __restrict__ agg, int e) {
  int t = blockIdx.x * blockDim.x + threadIdx.x;
  int eidx = t >> 4;
  if (eidx < e) {
    int f = t & 15;
    int s = src[eidx];
    int d = dst[eidx];
    float w = dinv[s] * dinv[d];
    atomicAdd(&agg[d * kHid + f], h[s * kHid + f] * w);
  }
}

// ---------------------------------------------------------------------------
// z[n] = relu(agg[n]+b1) . W2  ;  out[n] = b2 + z[n]*dinv[n]^2 (self loop)
// Each thread reads exactly its own 64B cacheline of agg via 4x b128 loads.
// ---------------------------------------------------------------------------
__global__ void k_z_out(const float* __restrict__ agg,
                        const float* __restrict__ dinv,
                        const float* __restrict__ b1,
                        const float* __restrict__ W2,
                        const float* __restrict__ b2,
                        float* __restrict__ z, float* __restrict__ out, int n) {
  int i = blockIdx.x * blockDim.x + threadIdx.x;
  if (i >= n) return;
  const float4* a4 = (const float4*)(agg + i * kHid);  // agg is 256B-aligned
  const float4* b4 = (const float4*)b1;
  const float4* w4 = (const float4*)W2;
  float acc = 0.0f;
#pragma unroll
  for (int q = 0; q < 4; ++q) {
    float4 av = a4[q], bv = b4[q], wv = w4[q];
    float v0 = av.x + bv.x; v0 = v0 > 0.0f ? v0 : 0.0f;
    float v1 = av.y + bv.y; v1 = v1 > 0.0f ? v1 : 0.0f;
    float v2 = av.z + bv.z; v2 = v2 > 0.0f ? v2 : 0.0f;
    float v3 = av.w + bv.w; v3 = v3 > 0.0f ? v3 : 0.0f;
    acc += v0 * wv.x + v1 * wv.y + v2 * wv.z + v3 * wv.w;
  }
  z[i] = acc;
  float di = dinv[i];
  out[i] = b2[0] + acc * di * di;
}

// ---------------------------------------------------------------------------
// Layer-2 edge pass (scalar): out[d] += z[s] * dinv[s]*dinv[d]
// ---------------------------------------------------------------------------
__global__ void k_edge1(const int* __restrict__ src, const int* __restrict__ dst,
                        const float* __restrict__ z, const float* __restrict__ dinv,
                        float* __restrict__ out, int e) {
  int i = blockIdx.x * blockDim.x + threadIdx.x;
  if (i < e) {
    int s = src[i];
    int d = dst[i];
    atomicAdd(&out[d], z[s] * dinv[s] * dinv[d]);
  }
}

extern "C" void kernel_launch(void* const* d_in, const int* in_sizes, int n_in,
                              void* d_out, int out_size, void* d_ws, size_t ws_size,
                              hipStream_t stream) {
  // setup_inputs order: x, edge_index, W1, b1, W2, b2
  const float* x  = (const float*)d_in[0];
  const int*   ei = (const int*)d_in[1];   // [2][E] int32 (JAX x64-off canonicalizes int64->int32)
  const float* W1 = (const float*)d_in[2];
  const float* b1 = (const float*)d_in[3];
  const float* W2 = (const float*)d_in[4];
  const float* b2 = (const float*)d_in[5];
  float* out = (float*)d_out;

  const int N = in_sizes[0] / kFin;   // 100000
  const int E = in_sizes[1] / 2;      // 3200000
  const int* src = ei;
  const int* dst = ei + E;

  // Workspace layout (floats), each partition 256B-aligned:
  //   dinv[N] | h[16N] | agg[16N] | z[N]   (~13.6 MB total)
  auto align64f = [](size_t v) { return (v + 63) & ~size_t(63); };  // 64 floats = 256B
  float* ws   = (float*)d_ws;
  size_t off  = 0;
  float* dinv = ws + off; off = align64f(off + (size_t)N);
  float* h    = ws + off; off = align64f(off + (size_t)kHid * N);
  float* agg  = ws + off; off = align64f(off + (size_t)kHid * N);
  float* z    = ws + off;

  const int B = 256;  // 8 wave32s per block
  const int ntiles = (N + 15) / 16;

  k_init_deg<<<(N + B - 1) / B, B, 0, stream>>>(dinv, N);
  k_degree<<<(E + B - 1) / B, B, 0, stream>>>(dst, dinv, E);
  k_rsqrt<<<(N + B - 1) / B, B, 0, stream>>>(dinv, N);

  k_xw1_wmma<<<(ntiles + 7) / 8, B, 0, stream>>>(x, W1, h, N, ntiles);

  k_selfloop16<<<(kHid * N + B - 1) / B, B, 0, stream>>>(h, dinv, agg, kHid * N);

  long long t16 = (long long)E * kHid;  // 51.2M threads
  k_edge16<<<(unsigned)((t16 + B - 1) / B), B, 0, stream>>>(src, dst, h, dinv, agg, E);

  k_z_out<<<(N + B - 1) / B, B, 0, stream>>>(agg, dinv, b1, W2, b2, z, out, N);

  k_edge1<<<(E + B - 1) / B, B, 0, stream>>>(src, dst, z, dinv, out, E);
}